// Tokenizer_50697793962313
// MI455X (gfx1250) — compile-verified
//
#include <hip/hip_runtime.h>

typedef __attribute__((ext_vector_type(2))) float v2f;
typedef __attribute__((ext_vector_type(8))) float v8f;

// ---- problem constants ----
#define BATCH      128
#define TOT_ROWS   340        // 4+16+64+256 feature rows per batch
#define FDIM       768        // 3*16*16
#define GATH_FLOATS (BATCH * 112 * FDIM)   // 11,010,048
#define IDS_ELEMS   (4 * BATCH * 256)      // 131,072

__constant__ __device__ const int c_featOff[4] = {0, 4, 20, 84};
__constant__ __device__ const int c_keepOff[4] = {0, 4, 16, 48};
__constant__ __device__ const int c_keepL[4]   = {4, 12, 32, 64};
__constant__ __device__ const int c_L[4]       = {4, 16, 64, 256};

// ---------------------------------------------------------------------------
// Build the jax.image.resize (linear, antialias) 16 x PS interpolation rows.
// ---------------------------------------------------------------------------
__global__ void tok_init_A(float* A0, float* A1, float* A2) {
    int t = threadIdx.x;
    if (t >= 48) return;
    int lvl = t >> 4, j = t & 15;
    int PS = 128 >> lvl;
    float* A = (lvl == 0) ? A0 : (lvl == 1) ? A1 : A2;
    float ks = (float)PS / 16.0f;                  // inv_scale == kernel_scale
    float sf = (j + 0.5f) * ks - 0.5f;
    float sum = 0.0f;
    for (int k = 0; k < PS; ++k) {
        float w = fmaxf(0.0f, 1.0f - fabsf(sf - (float)k) / ks);
        sum += w;
    }
    float inv = 1.0f / sum;
    for (int k = 0; k < PS; ++k) {
        float w = fmaxf(0.0f, 1.0f - fabsf(sf - (float)k) / ks);
        A[j * PS + k] = w * inv;
    }
}

// ---------------------------------------------------------------------------
// One wave per (batch, patch, channel): D(16x16) = A(16xPS) * P(PSxPS) * A^T.
// Uses V_WMMA_F32_16X16X4_F32 with T = A*P staged through LDS.
// ---------------------------------------------------------------------------
template <int PS, int PA, int LVL_OFF>
__global__ __launch_bounds__(32)
void tok_resize_wmma(const float* __restrict__ x, const float* __restrict__ A,
                     float* __restrict__ feats) {
    constexpr int L = PA * PA;
    __shared__ float T[16 * PS];

    int unit = blockIdx.x;                    // b*(L*3) + p*3 + c
    int c = unit % 3;
    int p = (unit / 3) % L;
    int b = unit / (3 * L);
    int pi = p / PA, pj = p % PA;

    const float* P = x + (((size_t)b * 3 + c) * 256 + (size_t)pi * PS) * 256 + (size_t)pj * PS;

    int lane = threadIdx.x;
    int half = lane >> 4;          // 0: lanes 0-15, 1: lanes 16-31
    int l15  = lane & 15;
    int kofs = 2 * half;           // K sub-offset per documented 16x4 f32 layout

    // ---- Phase 1: T = A * P  (tile over N, accumulate over K in steps of 4)
    for (int n = 0; n < PS; n += 16) {
        v8f acc = {};
        for (int k = 0; k < PS; k += 4) {
            v2f afrag;                                  // A[l15, k+kofs .. +1]
            afrag.x = A[l15 * PS + k + kofs];
            afrag.y = A[l15 * PS + k + kofs + 1];
            v2f bfrag;                                  // P[k+kofs .. +1, n+l15]
            int row = k + kofs;
            bfrag.x = P[(size_t)row * 256 + n + l15];
            bfrag.y = P[(size_t)(row + 1) * 256 + n + l15];
            acc = __builtin_amdgcn_wmma_f32_16x16x4_f32(
                false, afrag, false, bfrag, (short)0, acc, false, false);
        }
        // C/D layout: VGPR r -> M = r + 8*half, N = n + l15
        for (int r = 0; r < 8; ++r) {
            int M = r + 8 * half;
            T[M * PS + n + l15] = acc[r];
        }
    }
    __syncthreads();

    // ---- Phase 2: D = T * A^T  (single 16x16 output tile)
    v8f acc = {};
    for (int k = 0; k < PS; k += 4) {
        v2f afrag;                                      // T[l15, k+kofs .. +1]
        afrag.x = T[l15 * PS + k + kofs];
        afrag.y = T[l15 * PS + k + kofs + 1];
        v2f bfrag;                                      // (A^T)[k+kofs, l15] = A[l15, k+kofs]
        bfrag.x = A[l15 * PS + k + kofs];
        bfrag.y = A[l15 * PS + k + kofs + 1];
        acc = __builtin_amdgcn_wmma_f32_16x16x4_f32(
            false, afrag, false, bfrag, (short)0, acc, false, false);
    }

    float* F = feats + ((size_t)b * TOT_ROWS + LVL_OFF + p) * FDIM + c * 256;
    for (int r = 0; r < 8; ++r) {
        int M = r + 8 * half;
        F[M * 16 + l15] = acc[r];
    }
}

// ---------------------------------------------------------------------------
// Level 3 (ps=16): resize is the identity -> straight float4 copy.
// ---------------------------------------------------------------------------
__global__ void tok_copy_lvl3(const float* __restrict__ x, float* __restrict__ feats) {
    int g = blockIdx.x * blockDim.x + threadIdx.x;     // 6,291,456 float4 groups
    if (g >= BATCH * 256 * 3 * 64) return;
    int c4 = g & 3;
    int r  = (g >> 2) & 15;
    int t  = g >> 6;               // ((b*256 + p)*3 + c)
    int c  = t % 3;
    int p  = (t / 3) % 256;
    int b  = t / (3 * 256);
    int pi = p >> 4, pj = p & 15;

    const float4* src = (const float4*)(x + (((size_t)b * 3 + c) * 256 + pi * 16 + r) * 256
                                          + pj * 16 + c4 * 4);
    float4* dst = (float4*)(feats + ((size_t)b * TOT_ROWS + 84 + p) * FDIM
                                  + c * 256 + r * 16 + c4 * 4);
    *dst = *src;
}

// ---------------------------------------------------------------------------
// Entropy: one wave per feature row, -sum(f * log f).
// ---------------------------------------------------------------------------
__global__ void tok_entropy(const float* __restrict__ feats, float* __restrict__ ent) {
    int wid  = (blockIdx.x * blockDim.x + threadIdx.x) >> 5;
    int lane = threadIdx.x & 31;
    if (wid >= BATCH * TOT_ROWS) return;
    const float* f = feats + (size_t)wid * FDIM;
    float s = 0.0f;
    for (int i = lane; i < FDIM; i += 32) {
        float v = f[i];
        s += v * __logf(v);
    }
    #pragma unroll
    for (int off = 16; off > 0; off >>= 1) s += __shfl_down(s, off, 32);
    if (lane == 0) ent[wid] = -s;
}

// ---------------------------------------------------------------------------
// Stable rank == argsort(argsort) in one pass. Writes ids_restore (padded
// with zeros to 256) and scatters keep indices for the gather stage.
// ---------------------------------------------------------------------------
__global__ void tok_rank(const float* __restrict__ ent, int* __restrict__ keepidx,
                         void* __restrict__ ids_out, int ids_as_f32) {
    int gid = blockIdx.x * blockDim.x + threadIdx.x;   // 4 * 128 * 256
    if (gid >= IDS_ELEMS) return;
    int p   = gid & 255;
    int b   = (gid >> 8) & 127;
    int lvl = gid >> 15;
    int L   = c_L[lvl];
    long long rank = 0;
    if (p < L) {
        const float* e = ent + b * TOT_ROWS + c_featOff[lvl];
        float ep = e[p];
        int r = 0;
        for (int q = 0; q < L; ++q) {
            float eq = e[q];
            r += (eq < ep) || (eq == ep && q < p);
        }
        rank = r;
        if (r < c_keepL[lvl]) keepidx[b * 112 + c_keepOff[lvl] + r] = p;
    }
    size_t oidx = ((size_t)lvl * BATCH + b) * 256 + p;
    if (ids_as_f32) ((float*)ids_out)[oidx] = (float)rank;
    else            ((long long*)ids_out)[oidx] = rank;
}

// ---------------------------------------------------------------------------
// Gather kept feature rows: one wave per output row, float4 copies.
// ---------------------------------------------------------------------------
__global__ void tok_gather(const float* __restrict__ feats, const int* __restrict__ keepidx,
                           float* __restrict__ out) {
    int wid  = (blockIdx.x * blockDim.x + threadIdx.x) >> 5;
    int lane = threadIdx.x & 31;
    if (wid >= BATCH * 112) return;
    int b = wid / 112, j = wid % 112;
    int lvl = (j < 4) ? 0 : (j < 16) ? 1 : (j < 48) ? 2 : 3;
    int p = keepidx[wid];
    const float4* src = (const float4*)(feats + ((size_t)b * TOT_ROWS + c_featOff[lvl] + p) * FDIM);
    float4* dst = (float4*)(out + (size_t)wid * FDIM);
    for (int i = lane; i < FDIM / 4; i += 32) dst[i] = src[i];
}

// ---------------------------------------------------------------------------
extern "C" void kernel_launch(void* const* d_in, const int* in_sizes, int n_in,
                              void* d_out, int out_size, void* d_ws, size_t ws_size,
                              hipStream_t stream) {
    const float* x = (const float*)d_in[0];
    float* out = (float*)d_out;

    // workspace layout (floats)
    float* feats = (float*)d_ws;                               // 33,423,360
    float* ent   = feats + (size_t)BATCH * TOT_ROWS * FDIM;    // 43,520
    float* A0    = ent + BATCH * TOT_ROWS;                     // 16*128
    float* A1    = A0 + 16 * 128;                              // 16*64
    float* A2    = A1 + 16 * 64;                               // 16*32
    int*   keepidx = (int*)(A2 + 16 * 32);                     // 128*112 ints

    void* ids_out = (void*)(out + GATH_FLOATS);
    // int64 ids occupy 2 "floats" each; branch on what the harness allocated.
    int ids_as_f32 = ((size_t)out_size <= (size_t)(GATH_FLOATS + IDS_ELEMS)) ? 1 : 0;

    tok_init_A<<<1, 64, 0, stream>>>(A0, A1, A2);

    tok_resize_wmma<128, 2, 0 ><<<BATCH * 4  * 3, 32, 0, stream>>>(x, A0, feats);
    tok_resize_wmma<64,  4, 4 ><<<BATCH * 16 * 3, 32, 0, stream>>>(x, A1, feats);
    tok_resize_wmma<32,  8, 20><<<BATCH * 64 * 3, 32, 0, stream>>>(x, A2, feats);

    {
        int groups = BATCH * 256 * 3 * 64;
        tok_copy_lvl3<<<(groups + 255) / 256, 256, 0, stream>>>(x, feats);
    }
    {
        int threads = BATCH * TOT_ROWS * 32;
        tok_entropy<<<(threads + 255) / 256, 256, 0, stream>>>(feats, ent);
    }
    tok_rank<<<(IDS_ELEMS + 255) / 256, 256, 0, stream>>>(ent, keepidx, ids_out, ids_as_f32);
    {
        int threads = BATCH * 112 * 32;
        tok_gather<<<(threads + 255) / 256, 256, 0, stream>>>(feats, keepidx, out);
    }
}